// TECPredictor_56856777065092
// MI455X (gfx1250) — compile-verified
//
#include <hip/hip_runtime.h>
#include <hip/hip_bf16.h>

typedef __attribute__((ext_vector_type(2))) float v2f;
typedef __attribute__((ext_vector_type(8))) float v8f;

#define BATCH 128
#define TSEQ  512
#define HD    68   // padded LDS row stride (floats): 272B = 16B-aligned rows,
                   // bank = (4*row + col) % 64 -> conflict-free A-fragment loads
#define PS    17   // padded scratch stride (16 + 1)

__device__ __forceinline__ float fsigmoid(float x) {
    return 1.0f / (1.0f + __expf(-x));
}
__device__ __forceinline__ float ftanh(float x) {
    x = fminf(fmaxf(x, -15.0f), 15.0f);
    float e = __expf(2.0f * x);
    return (e - 1.0f) / (e + 1.0f);
}

// ---------------------------------------------------------------------------
// Kernel 1: bidirectional LSTM recurrence. One wave32 per (batch, direction).
// lane = hidden index. U (32x128) and W (16x128) columns live in VGPRs;
// h[j] broadcast via v_readlane (constant lane after unroll) -> SGPR FMAs.
// Writes hcat[b][t][dir*32 + lane].
// ---------------------------------------------------------------------------
__global__ void __launch_bounds__(256)
bilstm_kernel(const float* __restrict__ x,
              const float* __restrict__ Wf, const float* __restrict__ Uf,
              const float* __restrict__ bf,
              const float* __restrict__ Wb, const float* __restrict__ Ub,
              const float* __restrict__ bb_,
              float* __restrict__ hcat) {
    int wave = blockIdx.x * (blockDim.x >> 5) + (threadIdx.x >> 5); // 0..255
    int lane = threadIdx.x & 31;
    int dir  = wave >> 7;          // 0 = forward, 1 = backward
    int b    = wave & 127;

    const float* W    = dir ? Wb  : Wf;
    const float* U    = dir ? Ub  : Uf;
    const float* bias = dir ? bb_ : bf;

    // Recurrent weights: lane's 4 gate columns (i,f,g,o) for every j.
    float u[32][4];
    #pragma unroll
    for (int j = 0; j < 32; ++j)
        #pragma unroll
        for (int g = 0; g < 4; ++g)
            u[j][g] = U[j * 128 + g * 32 + lane];

    float w[16][4];
    #pragma unroll
    for (int d = 0; d < 16; ++d)
        #pragma unroll
        for (int g = 0; g < 4; ++g)
            w[d][g] = W[d * 128 + g * 32 + lane];

    float bz[4];
    #pragma unroll
    for (int g = 0; g < 4; ++g) bz[g] = bias[g * 32 + lane];

    float h = 0.0f, c = 0.0f;
    for (int t = 0; t < TSEQ; ++t) {
        int tin = dir ? (TSEQ - 1 - t) : t;
        const float* xr = x + ((size_t)b * TSEQ + tin) * 16;

        float z[4] = { bz[0], bz[1], bz[2], bz[3] };
        #pragma unroll
        for (int d = 0; d < 16; ++d) {
            float xd = xr[d];                       // wave-uniform -> scalar load
            #pragma unroll
            for (int g = 0; g < 4; ++g) z[g] += xd * w[d][g];
        }
        #pragma unroll
        for (int j = 0; j < 32; ++j) {
            float hj = __int_as_float(
                __builtin_amdgcn_readlane(__float_as_int(h), j));
            #pragma unroll
            for (int g = 0; g < 4; ++g) z[g] += hj * u[j][g];
        }
        float ig = fsigmoid(z[0]);
        float fg = fsigmoid(z[1]);
        float gg = ftanh(z[2]);
        float og = fsigmoid(z[3]);
        c = fg * c + ig * gg;
        h = og * ftanh(c);
        hcat[((size_t)b * TSEQ + tin) * 64 + dir * 32 + lane] = h;
    }
}

// ---------------------------------------------------------------------------
// Kernel 2: per-batch self-attention + pooling, WMMA f32 16x16x4.
// h[b] (512x64) loaded LDS-resident via async global->LDS (B128, padded rows).
// Fixed-shift softmax (scores bounded in [-64,64], diagonal >= 0 guarantees
// the row max is >= 0, so exp(s-64) never collapses a whole row): no
// cross-lane reductions at all — row sums computed by WMMA against ones.
// All B-fragments are preloaded into registers ahead of each WMMA chain so
// LDS latency overlaps instead of serializing (s_wait_dscnt 0 per WMMA).
// ---------------------------------------------------------------------------
__global__ void __launch_bounds__(512)
attn_pool_kernel(const float* __restrict__ hcat, float* __restrict__ pooled) {
    extern __shared__ float smem[];
    float* hl  = smem;                      // 512 * 68
    float* scr = smem + TSEQ * HD;          // 16 waves * 16 * 17
    float* pl  = scr + 16 * 16 * PS;        // 64

    int b    = blockIdx.x;
    int tid  = threadIdx.x;
    int wave = tid >> 5;
    int lane = tid & 31;
    int m16  = lane & 15;
    int hi   = lane >> 4;                   // 0: lanes 0-15, 1: lanes 16-31

    // --- async global -> LDS load of h[b], scattered into padded rows ------
    {
        const float* hg = hcat + (size_t)b * (TSEQ * 64);
        unsigned gss = __builtin_amdgcn_groupstaticsize(); // dynamic-LDS base
        #pragma unroll 4
        for (int i = 0; i < 16; ++i) {
            int cidx = tid + i * 512;                 // 16-byte chunk index
            int row  = cidx >> 4;                     // 16 chunks per row
            int cin  = cidx & 15;
            unsigned ldsoff = gss + (unsigned)(row * HD + cin * 4) * 4u;
            const float* gp = hg + (size_t)cidx * 4;
            asm volatile("global_load_async_to_lds_b128 %0, %1, off"
                         :: "v"(ldsoff), "v"(gp) : "memory");
        }
        asm volatile("s_wait_asynccnt 0x0" ::: "memory");
    }
    if (tid < 64) pl[tid] = 0.0f;
    __syncthreads();

    float* ws = scr + wave * (16 * PS);
    const float SHIFT = 64.0f;
    v2f onesv; onesv[0] = 1.0f; onesv[1] = 1.0f;

    for (int qi = 0; qi < 2; ++qi) {
        int qt = wave + qi * 16;            // 0..31
        int q0 = qt * 16;

        // Preload Q fragments in WMMA A-layout (reused across all key tiles).
        float qa0[16], qa1[16];
        #pragma unroll
        for (int k4 = 0; k4 < 16; ++k4) {
            qa0[k4] = hl[(q0 + m16) * HD + k4 * 4 + 2 * hi];
            qa1[k4] = hl[(q0 + m16) * HD + k4 * 4 + 2 * hi + 1];
        }

        v8f acc[4];                         // att numerator, 4 col tiles
        v8f lacc;                           // softmax denominator (row sums)
        #pragma unroll
        for (int ct = 0; ct < 4; ++ct)
            #pragma unroll
            for (int r = 0; r < 8; ++r) acc[ct][r] = 0.0f;
        #pragma unroll
        for (int r = 0; r < 8; ++r) lacc[r] = 0.0f;

        for (int kt = 0; kt < 32; ++kt) {
            int k0 = kt * 16;

            // Preload ALL K fragments for this tile (keeps 16 loads in
            // flight; WMMA chain then runs without per-op dscnt stalls).
            float kb0[16], kb1[16];
            #pragma unroll
            for (int k4 = 0; k4 < 16; ++k4) {
                kb0[k4] = hl[(k0 + m16) * HD + k4 * 4 + 2 * hi];
                kb1[k4] = hl[(k0 + m16) * HD + k4 * 4 + 2 * hi + 1];
            }

            // S tile (16x16) = Q(16x64) @ K^T : chain of 16 K=4 WMMAs.
            v8f s;
            #pragma unroll
            for (int r = 0; r < 8; ++r) s[r] = 0.0f;
            #pragma unroll
            for (int k4 = 0; k4 < 16; ++k4) {
                v2f a;  a[0] = qa0[k4];  a[1] = qa1[k4];
                v2f bv; bv[0] = kb0[k4]; bv[1] = kb1[k4];
                s = __builtin_amdgcn_wmma_f32_16x16x4_f32(
                        false, a, false, bv, (short)0, s, false, false);
            }

            // Preload ALL V fragments now: their LDS latency hides under the
            // exp() evaluation and the P scratch round-trip below.
            float vb0[16], vb1[16];            // [ct*4 + k4]
            #pragma unroll
            for (int ct = 0; ct < 4; ++ct)
                #pragma unroll
                for (int k4 = 0; k4 < 4; ++k4) {
                    vb0[ct * 4 + k4] =
                        hl[(k0 + k4 * 4 + 2 * hi)     * HD + ct * 16 + m16];
                    vb1[ct * 4 + k4] =
                        hl[(k0 + k4 * 4 + 2 * hi + 1) * HD + ct * 16 + m16];
                }

            // P = exp(S - 64): safe fixed shift, no row-max reduction needed.
            #pragma unroll
            for (int r = 0; r < 8; ++r) s[r] = __expf(s[r] - SHIFT);

            // C/D layout -> A layout via wave-private LDS scratch.
            #pragma unroll
            for (int r = 0; r < 8; ++r)
                ws[(r + 8 * hi) * PS + m16] = s[r];
            float pa0[4], pa1[4];
            #pragma unroll
            for (int k4 = 0; k4 < 4; ++k4) {
                pa0[k4] = ws[m16 * PS + k4 * 4 + 2 * hi];
                pa1[k4] = ws[m16 * PS + k4 * 4 + 2 * hi + 1];
            }

            // Row sums: lacc += P @ ones  (lands in C/D layout, every lane).
            #pragma unroll
            for (int k4 = 0; k4 < 4; ++k4) {
                v2f a;  a[0] = pa0[k4];  a[1] = pa1[k4];
                lacc = __builtin_amdgcn_wmma_f32_16x16x4_f32(
                           false, a, false, onesv, (short)0, lacc, false, false);
            }

            // acc += P(16x16) @ V(16x64): 4 col tiles x 4 K=4 WMMAs.
            #pragma unroll
            for (int ct = 0; ct < 4; ++ct) {
                #pragma unroll
                for (int k4 = 0; k4 < 4; ++k4) {
                    v2f a;  a[0] = pa0[k4];       a[1] = pa1[k4];
                    v2f bv; bv[0] = vb0[ct * 4 + k4]; bv[1] = vb1[ct * 4 + k4];
                    acc[ct] = __builtin_amdgcn_wmma_f32_16x16x4_f32(
                                  false, a, false, bv, (short)0, acc[ct], false, false);
                }
            }
        }

        // att = acc / l ; pooled[d] += sum over this tile's rows of h*att.
        float invl[8];
        #pragma unroll
        for (int r = 0; r < 8; ++r) invl[r] = 1.0f / lacc[r];
        #pragma unroll
        for (int ct = 0; ct < 4; ++ct) {
            int d = ct * 16 + m16;
            float sacc = 0.0f;
            #pragma unroll
            for (int r = 0; r < 8; ++r) {
                float att = acc[ct][r] * invl[r];
                sacc += hl[(q0 + r + 8 * hi) * HD + d] * att;
            }
            atomicAdd(&pl[d], sacc);            // ds_add_f32
        }
    }
    __syncthreads();
    if (tid < 64) pooled[b * 64 + tid] = pl[tid];
}

// ---------------------------------------------------------------------------
// Kernel 3: second LSTM cell (zero state -> U2 term vanishes) + dense head.
// One block of 128 threads per batch element.
// ---------------------------------------------------------------------------
__global__ void __launch_bounds__(128)
head_kernel(const float* __restrict__ pooled,
            const float* __restrict__ W2,  const float* __restrict__ b2,
            const float* __restrict__ Wd1, const float* __restrict__ bd1,
            const float* __restrict__ Wd2, const float* __restrict__ bd2,
            float* __restrict__ out) {
    __shared__ float zs[128];
    __shared__ float h2[32];
    __shared__ float y1[16];
    int b = blockIdx.x, k = threadIdx.x;
    const float* p = pooled + b * 64;

    float z = b2[k];
    #pragma unroll 8
    for (int d = 0; d < 64; ++d) z += p[d] * W2[d * 128 + k];
    zs[k] = z;
    __syncthreads();

    if (k < 32) {
        float cc = fsigmoid(zs[k]) * ftanh(zs[64 + k]);   // i * tanh(g)
        h2[k] = fsigmoid(zs[96 + k]) * ftanh(cc);         // o * tanh(c)
    }
    __syncthreads();

    if (k < 16) {
        float a = bd1[k];
        #pragma unroll
        for (int j = 0; j < 32; ++j) a += h2[j] * Wd1[j * 16 + k];
        y1[k] = fmaxf(a, 0.0f);
    }
    __syncthreads();

    if (k == 0) {
        float a = bd2[0];
        #pragma unroll
        for (int j = 0; j < 16; ++j) a += y1[j] * Wd2[j];
        out[b] = a;
    }
}

extern "C" void kernel_launch(void* const* d_in, const int* in_sizes, int n_in,
                              void* d_out, int out_size, void* d_ws, size_t ws_size,
                              hipStream_t stream) {
    (void)in_sizes; (void)n_in; (void)out_size; (void)ws_size;
    const float* x   = (const float*)d_in[0];
    const float* Wf  = (const float*)d_in[1];
    const float* Uf  = (const float*)d_in[2];
    const float* bf  = (const float*)d_in[3];
    const float* Wb  = (const float*)d_in[4];
    const float* Ub  = (const float*)d_in[5];
    const float* bb  = (const float*)d_in[6];
    const float* W2  = (const float*)d_in[7];
    /* d_in[8] = U2: unused, zero initial state makes its term vanish */
    const float* b2  = (const float*)d_in[9];
    const float* Wd1 = (const float*)d_in[10];
    const float* bd1 = (const float*)d_in[11];
    const float* Wd2 = (const float*)d_in[12];
    const float* bd2 = (const float*)d_in[13];
    float* out = (float*)d_out;

    float* hcat   = (float*)d_ws;                          // 128*512*64 f32 = 16 MB
    float* pooled = hcat + (size_t)BATCH * TSEQ * 64;      // 128*64 f32

    bilstm_kernel<<<32, 256, 0, stream>>>(x, Wf, Uf, bf, Wb, Ub, bb, hcat);

    size_t smem = (size_t)(TSEQ * HD + 16 * 16 * PS + 64) * sizeof(float); // ~153 KB
    attn_pool_kernel<<<BATCH, 512, smem, stream>>>(hcat, pooled);

    head_kernel<<<BATCH, 128, 0, stream>>>(pooled, W2, b2, Wd1, bd1, Wd2, bd2, out);
}